// LDDMMVariational_43731357008441
// MI455X (gfx1250) — compile-verified
//
#include <hip/hip_runtime.h>

// LDDMM variational evolve on MI455X (gfx1250, wave32).
//
// Compute-bound O(N^2) Gaussian N-body (N=8192, D=3, fp32; ~10^4 flops/byte).
// Per 16x16 (i,j) tile, two V_WMMA_F32_16X16X4_F32 ops produce
//   d2[m][n] = ||x_m||^2 + ||x_n||^2 - 2 x_m.x_n  (norms in K=4 slot + C input)
//   cm[m][n] = mom_m . mom_n
// then per-lane VALU does K = v_exp_f32(-50*log2e*d2), W = K*cm and packed-FMA
// accumulation of
//   dpos_i += K * mom_j
//   dmom_i  = 100 * ( x_i * sum_j W - sum_j W x_j )
// C/D lane layout: lane = one j column, 8 i rows per wave half. dmom is folded
// per-lane BEFORE the 4-step shfl_xor butterfly (48 values) to shrink the ds
// reduction; j split 8 ways across waves (4096 waves) to fill the machine,
// partials merged with global_atomic_add_f32 into a memset-zeroed output.

typedef __attribute__((ext_vector_type(2))) float v2f;
typedef __attribute__((ext_vector_type(8))) float v8f;

#define N_PTS 8192
#define TILE 16
#define JCHUNKS 8
#define NTILES (N_PTS / TILE)                   // 512 j-tiles total
#define TILES_PER_CHUNK (NTILES / JCHUNKS)      // 64 j-tiles per wave
#define WAVES_PER_BLOCK 8
#define TOTAL_WAVES (NTILES * JCHUNKS)          // 4096 waves
#define NBLOCKS (TOTAL_WAVES / WAVES_PER_BLOCK) // 512 blocks

// exp(-d2/(2*sigma^2)) = exp2(-50*log2(e)*d2), sigma = 0.1
#define NEG_50_LOG2E (-72.13475204444817f)
#define INV_SIG2 (100.0f)

__device__ __forceinline__ void atomic_add_f32(float* p, float v) {
    __hip_atomic_fetch_add(p, v, __ATOMIC_RELAXED, __HIP_MEMORY_SCOPE_AGENT);
}

__global__ __launch_bounds__(256) void lddmm_variational_kernel(
    const float* __restrict__ mom, const float* __restrict__ pos,
    float* __restrict__ dmom, float* __restrict__ dpos)
{
    const int lane        = threadIdx.x & 31;
    const int wave_in_blk = threadIdx.x >> 5;
    const int wave        = blockIdx.x * WAVES_PER_BLOCK + wave_in_blk;
    const int tile_i      = wave / JCHUNKS;   // which 16-row i tile
    const int chunk       = wave % JCHUNKS;   // which j chunk
    const int i0          = tile_i * TILE;

    const int n = lane & 15;   // A-layout: M row index; B/C/D-layout: j column
    const int h = lane >> 4;   // wave half

    // ---------- A matrices (fixed for the whole j loop) ----------
    // A layout (16x4 fp32): lane holds row M=lane%16; comp v, half h -> K = v + 2h.
    const int mrow = i0 + n;
    const float p0 = pos[3 * mrow + 0], p1 = pos[3 * mrow + 1], p2 = pos[3 * mrow + 2];
    const float q0 = mom[3 * mrow + 0], q1 = mom[3 * mrow + 1], q2 = mom[3 * mrow + 2];

    v2f a_pos, a_mom;
    a_pos.x = h ? (-2.0f * p2) : (-2.0f * p0);  // K=2h   : -2*x[m][2h]
    a_pos.y = h ? 1.0f         : (-2.0f * p1);  // K=3 slot carries "1" for +||x_j||^2
    a_mom.x = h ? q2   : q0;
    a_mom.y = h ? 0.0f : q1;

    // ---------- broadcasts: row norms (C input) and pos rows (for dmom fold) ----------
    // D/C layout: comp r, half h -> row m = r + 8h; lane L%16 holds row L%16 data.
    const float nxi = p0 * p0 + p1 * p1 + p2 * p2;
    v8f c_nx;
    float pm[24];   // pm[3r+d] = pos[i0 + r + 8h][d], lane-invariant per half
#pragma unroll
    for (int r = 0; r < 8; ++r) {
        const int src = r + 8 * h;
        c_nx[r]       = __shfl(nxi, src, 32);
        pm[3 * r + 0] = __shfl(p0,  src, 32);
        pm[3 * r + 1] = __shfl(p1,  src, 32);
        pm[3 * r + 2] = __shfl(p2,  src, 32);
    }

    // ---------- accumulators (per lane: fixed j column, 8 i rows) ----------
    float acc_dpos[24], acc_wp[24], acc_s[8];
#pragma unroll
    for (int r = 0; r < 8; ++r) {
        acc_s[r] = 0.0f;
#pragma unroll
        for (int d = 0; d < 3; ++d) {
            acc_dpos[3 * r + d] = 0.0f;
            acc_wp[3 * r + d]   = 0.0f;
        }
    }

    // ---------- j-tile loop ----------
    const int jt0 = chunk * TILES_PER_CHUNK;
    for (int jt = 0; jt < TILES_PER_CHUNK; ++jt) {
        const int j = (jt0 + jt) * TILE + n;
        const float xj0 = pos[3 * j + 0], xj1 = pos[3 * j + 1], xj2 = pos[3 * j + 2];
        const float yj0 = mom[3 * j + 0], yj1 = mom[3 * j + 1], yj2 = mom[3 * j + 2];
        const float nxj = xj0 * xj0 + xj1 * xj1 + xj2 * xj2;

        // B layout (4x16 fp32): lane holds column N=lane%16; comp v, half h -> K = v + 2h.
        v2f b_pos, b_mom;
        b_pos.x = h ? xj2 : xj0;
        b_pos.y = h ? nxj : xj1;   // K=3 row carries ||x_j||^2
        b_mom.x = h ? yj2 : yj0;
        b_mom.y = h ? 0.0f : yj1;

        // d2[m][n] = -2 x_m.x_n + ||x_n||^2 + ||x_m||^2
        v8f d2 = __builtin_amdgcn_wmma_f32_16x16x4_f32(
            false, a_pos, false, b_pos, (short)0, c_nx, false, false);
        // cm[m][n] = mom_m . mom_n
        v8f cz = {};
        v8f cm = __builtin_amdgcn_wmma_f32_16x16x4_f32(
            false, a_mom, false, b_mom, (short)0, cz, false, false);

#pragma unroll
        for (int r = 0; r < 8; ++r) {
            const float K = __builtin_amdgcn_exp2f(NEG_50_LOG2E * d2[r]); // v_exp_f32
            const float W = K * cm[r];
            acc_dpos[3 * r + 0] += K * yj0;
            acc_dpos[3 * r + 1] += K * yj1;
            acc_dpos[3 * r + 2] += K * yj2;
            acc_s[r] += W;
            acc_wp[3 * r + 0] += W * xj0;
            acc_wp[3 * r + 1] += W * xj1;
            acc_wp[3 * r + 2] += W * xj2;
        }
    }

    // ---------- fold dmom per-lane, then butterfly reduce 48 values ----------
    float dm[24];
#pragma unroll
    for (int r = 0; r < 8; ++r)
#pragma unroll
        for (int d = 0; d < 3; ++d)
            dm[3 * r + d] = INV_SIG2 * (pm[3 * r + d] * acc_s[r] - acc_wp[3 * r + d]);

#pragma unroll
    for (int m = 8; m >= 1; m >>= 1) {
#pragma unroll
        for (int v = 0; v < 24; ++v) {
            dm[v]       += __shfl_xor(dm[v],       m, 32);
            acc_dpos[v] += __shfl_xor(acc_dpos[v], m, 32);
        }
    }

    // ---------- writeback: lane n==r writes row m = r + 8h (6 atomics) ----------
    if (n < 8) {
        const int r = n;
        const int m = i0 + r + 8 * h;
        atomic_add_f32(&dmom[3 * m + 0], dm[3 * r + 0]);
        atomic_add_f32(&dmom[3 * m + 1], dm[3 * r + 1]);
        atomic_add_f32(&dmom[3 * m + 2], dm[3 * r + 2]);
        atomic_add_f32(&dpos[3 * m + 0], acc_dpos[3 * r + 0]);
        atomic_add_f32(&dpos[3 * m + 1], acc_dpos[3 * r + 1]);
        atomic_add_f32(&dpos[3 * m + 2], acc_dpos[3 * r + 2]);
    }
}

extern "C" void kernel_launch(void* const* d_in, const int* in_sizes, int n_in,
                              void* d_out, int out_size, void* d_ws, size_t ws_size,
                              hipStream_t stream) {
    (void)in_sizes; (void)n_in; (void)d_ws; (void)ws_size;
    const float* mom = (const float*)d_in[0];
    const float* pos = (const float*)d_in[1];
    float* out  = (float*)d_out;
    float* dmom = out;              // reference returns (dmom, dpos) concatenated
    float* dpos = out + N_PTS * 3;

    hipMemsetAsync(d_out, 0, (size_t)out_size * sizeof(float), stream);
    lddmm_variational_kernel<<<NBLOCKS, 256, 0, stream>>>(mom, pos, dmom, dpos);
}